// SelfAttentionPooling_22299470201569
// MI455X (gfx1250) — compile-verified
//
#include <hip/hip_runtime.h>
#include <math.h>

// CDNA5 fp32 WMMA operand types
typedef __attribute__((ext_vector_type(2))) float v2f;   // A (16x4) / B (4x16): 2 VGPRs
typedef __attribute__((ext_vector_type(8))) float v8f;   // C/D (16x16): 8 VGPRs

#define D_MODEL 512
#define T_SEQ   8192
#define B_BATCH 32
#define CHUNK   512                   // T rows per block
#define NCHUNK  (T_SEQ / CHUNK)       // 16 chunks per batch
#define STILE   128                   // rows per super-tile (8 waves x 16 rows)
#define NST     (CHUNK / STILE)       // 4 super-tiles per block

__device__ __forceinline__ float wave_max_f(float v) {
#pragma unroll
    for (int off = 16; off > 0; off >>= 1)
        v = fmaxf(v, __shfl_xor(v, off, 32));
    return v;
}
__device__ __forceinline__ float wave_sum_f(float v) {
#pragma unroll
    for (int off = 16; off > 0; off >>= 1)
        v += __shfl_xor(v, off, 32);
    return v;
}

// One block = (batch b, T-chunk). Single pass over its slice of x:
// computes scores with fp32 WMMA, online softmax, and WMMA-weighted
// accumulation. Emits unnormalized partial out + (m, l) per chunk.
__global__ __launch_bounds__(256)
void attnpool_fused(const float* __restrict__ x,
                    const float* __restrict__ w,
                    float* __restrict__ partial,   // [B*NCHUNK, D_MODEL]
                    float* __restrict__ m_arr,     // [B*NCHUNK]
                    float* __restrict__ l_arr)     // [B*NCHUNK]
{
    __shared__ float s_scores[STILE];
    __shared__ float s_p[STILE];

    const int b     = blockIdx.y;
    const int chunk = blockIdx.x;
    const int tid   = threadIdx.x;
    const int wave  = tid >> 5;      // 8 waves
    const int lane  = tid & 31;
    const int half  = lane >> 4;     // 0: lanes 0-15, 1: lanes 16-31
    const int lrow  = lane & 15;

    const size_t rowBase = (size_t)b * T_SEQ + (size_t)chunk * CHUNK;
    const int    dwave   = wave * 64;          // this wave's 64 output columns

    float m = -INFINITY;                       // running max
    float l = 0.0f;                            // running sum of exp
    v8f acc0 = {}, acc1 = {}, acc2 = {}, acc3 = {};

    for (int st = 0; st < NST; ++st) {
        __syncthreads();   // protect s_scores/s_p reuse across super-tiles

        // ---- Phase 1: each wave computes 16 scores via V_WMMA_F32_16X16X4_F32
        // A: 16 rows of x, K=4 slice. lanes 0-15 carry K=0,1; lanes 16-31 K=2,3.
        // B: w[d0+k] replicated across all 16 columns.
        const size_t r0 = rowBase + (size_t)st * STILE + (size_t)wave * 16;
        {
            const float* xr = x + (r0 + (size_t)lrow) * D_MODEL + 2 * half;
            const float* wr = w + 2 * half;
            v8f cs = {};
#pragma unroll 4
            for (int d0 = 0; d0 < D_MODEL; d0 += 4) {
                v2f a  = { xr[d0], xr[d0 + 1] };
                v2f bb = { wr[d0], wr[d0 + 1] };
                cs = __builtin_amdgcn_wmma_f32_16x16x4_f32(
                        false, a, false, bb, (short)0, cs, false, false);
            }
            // C row m lives in VGPR (m&7): lanes 0-15 hold M=0..7, 16-31 hold M=8..15
            if (lane == 0) {
#pragma unroll
                for (int i = 0; i < 8; ++i) s_scores[wave * 16 + i] = cs[i];
            }
            if (lane == 16) {
#pragma unroll
                for (int i = 0; i < 8; ++i) s_scores[wave * 16 + 8 + i] = cs[i];
            }
        }
        __syncthreads();

        // ---- Phase 2: online softmax update (identical in every wave -> deterministic)
        float s0 = s_scores[lane      ], s1 = s_scores[lane + 32];
        float s2 = s_scores[lane + 64 ], s3 = s_scores[lane + 96];
        float tmax  = wave_max_f(fmaxf(fmaxf(s0, s1), fmaxf(s2, s3)));
        float m_new = fmaxf(m, tmax);
        float f     = __expf(m - m_new);
        float esum  = __expf(s0 - m_new) + __expf(s1 - m_new) +
                      __expf(s2 - m_new) + __expf(s3 - m_new);
        esum = wave_sum_f(esum);
        l = l * f + esum;
        m = m_new;
#pragma unroll
        for (int i = 0; i < 8; ++i) {
            acc0[i] *= f; acc1[i] *= f; acc2[i] *= f; acc3[i] *= f;
        }
        if (lane < 16)   // wave w publishes p for its own 16 rows
            s_p[wave * 16 + lane] = __expf(s_scores[wave * 16 + lane] - m_new);
        __syncthreads();

        // ---- Phase 3: acc += p-tile x x-tile  (A = p replicated across 16 rows,
        // B[k,n] = x[rt+t+k, d0+n]; lanes 0-15 carry K=0,1 of column n, 16-31 K=2,3)
        const size_t rtile = rowBase + (size_t)st * STILE;
        for (int t = 0; t < STILE; t += 4) {
            v2f a = { s_p[t + 2 * half], s_p[t + 2 * half + 1] };
            const float* xb = x + (rtile + (size_t)t + 2 * half) * D_MODEL + dwave + lrow;
            v2f b0 = { xb[ 0], xb[ 0 + D_MODEL] };
            v2f b1 = { xb[16], xb[16 + D_MODEL] };
            v2f b2 = { xb[32], xb[32 + D_MODEL] };
            v2f b3 = { xb[48], xb[48 + D_MODEL] };
            acc0 = __builtin_amdgcn_wmma_f32_16x16x4_f32(false, a, false, b0, (short)0, acc0, false, false);
            acc1 = __builtin_amdgcn_wmma_f32_16x16x4_f32(false, a, false, b1, (short)0, acc1, false, false);
            acc2 = __builtin_amdgcn_wmma_f32_16x16x4_f32(false, a, false, b2, (short)0, acc2, false, false);
            acc3 = __builtin_amdgcn_wmma_f32_16x16x4_f32(false, a, false, b3, (short)0, acc3, false, false);
        }
    }

    // ---- Emit: C row 0 (M=0, N=lane for lanes 0-15) = partial out column d0+lane
    const int pbase = (b * NCHUNK + chunk) * D_MODEL;
    if (lane < 16) {
        partial[pbase + dwave +  0 + lane] = acc0[0];
        partial[pbase + dwave + 16 + lane] = acc1[0];
        partial[pbase + dwave + 32 + lane] = acc2[0];
        partial[pbase + dwave + 48 + lane] = acc3[0];
    }
    if (tid == 0) {
        m_arr[b * NCHUNK + chunk] = m;
        l_arr[b * NCHUNK + chunk] = l;
    }
}

// Deterministic fixed-order merge of the NCHUNK partials per batch.
__global__ __launch_bounds__(256)
void attnpool_merge(const float* __restrict__ partial,
                    const float* __restrict__ m_arr,
                    const float* __restrict__ l_arr,
                    float* __restrict__ out)
{
    const int b   = blockIdx.x;
    const int tid = threadIdx.x;

    float mc[NCHUNK], wc[NCHUNK];
    float M = -INFINITY;
#pragma unroll
    for (int c = 0; c < NCHUNK; ++c) {
        mc[c] = m_arr[b * NCHUNK + c];
        M = fmaxf(M, mc[c]);
    }
    float L = 0.0f;
#pragma unroll
    for (int c = 0; c < NCHUNK; ++c) {
        wc[c] = __expf(mc[c] - M);
        L += l_arr[b * NCHUNK + c] * wc[c];
    }
    const float invL = 1.0f / L;

    for (int d = tid; d < D_MODEL; d += 256) {
        float s = 0.0f;
#pragma unroll
        for (int c = 0; c < NCHUNK; ++c)
            s += wc[c] * partial[(b * NCHUNK + c) * D_MODEL + d];
        out[(size_t)b * D_MODEL + d] = s * invL;
    }
}

extern "C" void kernel_launch(void* const* d_in, const int* in_sizes, int n_in,
                              void* d_out, int out_size, void* d_ws, size_t ws_size,
                              hipStream_t stream) {
    (void)in_sizes; (void)n_in; (void)out_size; (void)ws_size;
    const float* x = (const float*)d_in[0];
    const float* w = (const float*)d_in[1];
    // d_in[2] is the bias; softmax is shift-invariant so it cannot affect out.

    float* ws      = (float*)d_ws;
    float* partial = ws;                                        // B*NCHUNK*D_MODEL floats (1 MiB)
    float* m_arr   = partial + B_BATCH * NCHUNK * D_MODEL;      // B*NCHUNK floats
    float* l_arr   = m_arr + B_BATCH * NCHUNK;                  // B*NCHUNK floats

    dim3 grid(NCHUNK, B_BATCH);   // 512 blocks, 256 threads (8 wave32 each)
    attnpool_fused<<<grid, 256, 0, stream>>>(x, w, partial, m_arr, l_arr);
    attnpool_merge<<<B_BATCH, 256, 0, stream>>>(partial, m_arr, l_arr, (float*)d_out);
}